// QuantizationLayer_1580547965017
// MI455X (gfx1250) — compile-verified
//
#include <hip/hip_runtime.h>
#include <hip/hip_bf16.h>

// Quantize-and-unpack: out[i*B + j] = bit (B-1-j) of rint(x[i]*2^B - 0.5), as f32.
// Pure streaming 1->B expansion; memory-bound (320 MB @ 23.3 TB/s ~= 13.7 us floor).
// No matrix structure -> no WMMA by design. gfx1250 feature used: non-temporal
// (TH_STORE_NT) stores so the 256 MB write stream doesn't thrash the 192 MB L2.

typedef __attribute__((ext_vector_type(4))) float v4f;

__global__ __launch_bounds__(256) void quant_unpack_b4(
    const float* __restrict__ x, v4f* __restrict__ out, int n) {
  int i = blockIdx.x * blockDim.x + threadIdx.x;
  if (i >= n) return;

  // coalesced 4B/lane load (128B per wave32)
  float v = x[i];

  // code = round-half-even(x*16 - 0.5); x*16 is exact, fma == separate mul/sub
  float r = __builtin_rintf(__builtin_fmaf(v, 16.0f, -0.5f));
  unsigned code = (unsigned)(int)r;  // in [0,15]; -0.0 -> 0

  // MSB-first bit unpack to floats (v_and + v_cndmask with inline 1.0)
  v4f b;
  b.x = (code & 8u) ? 1.0f : 0.0f;
  b.y = (code & 4u) ? 1.0f : 0.0f;
  b.z = (code & 2u) ? 1.0f : 0.0f;
  b.w = (code & 1u) ? 1.0f : 0.0f;

  // contiguous 16B/lane (512B per wave32) non-temporal store:
  // global_store_b128 ... th:TH_STORE_NT — streams past L2
  __builtin_nontemporal_store(b, out + i);
}

// Fallback for any B (scalar stores), selected host-side if B != 4.
__global__ __launch_bounds__(256) void quant_unpack_gen(
    const float* __restrict__ x, float* __restrict__ out, int n, int B) {
  int i = blockIdx.x * blockDim.x + threadIdx.x;
  if (i >= n) return;
  float step = (float)(1u << B);
  unsigned code = (unsigned)(int)__builtin_rintf(__builtin_fmaf(x[i], step, -0.5f));
  long base = (long)i * B;
  for (int j = 0; j < B; ++j) {
    float bit = (float)((code >> (B - 1 - j)) & 1u);
    __builtin_nontemporal_store(bit, out + base + j);
  }
}

extern "C" void kernel_launch(void* const* d_in, const int* in_sizes, int n_in,
                              void* d_out, int out_size, void* d_ws, size_t ws_size,
                              hipStream_t stream) {
  const float* x = (const float*)d_in[0];
  float* out = (float*)d_out;

  int n = in_sizes[0];          // 32768*512 = 16,777,216 input elements
  int B = out_size / n;         // expansion factor; == 4 for the reference setup

  const int threads = 256;      // 8 wave32 per block
  int blocks = (n + threads - 1) / threads;

  if (B == 4) {
    quant_unpack_b4<<<blocks, threads, 0, stream>>>(x, (v4f*)out, n);
  } else {
    quant_unpack_gen<<<blocks, threads, 0, stream>>>(x, out, n, B);
  }
}